// RNN_69724499083455
// MI455X (gfx1250) — compile-verified
//
#include <hip/hip_runtime.h>
#include <cmath>

typedef __attribute__((ext_vector_type(2))) float v2f;
typedef __attribute__((ext_vector_type(8))) float v8f;

#define T_DIM 1024
#define B_DIM 256
#define I_DIM 128
#define H_DIM 256

// Padded LDS strides: stride mod 64 = 4 -> 16-lane column reads hit 16 banks.
#define LDA 132   // 128 + 4
#define LDH 260   // 256 + 4

// Fast tanh: gfx1250 has a hardware TANH transcendental (TRANS32 op list).
__device__ __forceinline__ float fast_tanh(float x) {
#if __has_builtin(__builtin_amdgcn_tanhf)
  return __builtin_amdgcn_tanhf(x);
#elif __has_builtin(__builtin_amdgcn_tanh_f32)
  return __builtin_amdgcn_tanh_f32(x);
#else
  // tanh(x) = 1 - 2/(exp2(2x*log2 e) + 1);  saturates correctly at +/-1.
  const float e = __builtin_amdgcn_exp2f(x * 2.885390081777927f);
  return 1.0f - 2.0f / (e + 1.0f);
#endif
}

// -------------------------------------------------------------------------
// Kernel 1: xW[t,b,h] = sum_i x[t,b,i] * W[h,i] + b_W[h]
// M = T*B rows; each block does a 16-row M-tile x full N=256.
// 8 waves / block, each wave owns two 16-wide N subtiles (compiler
// interleaves the two independent WMMA chains for ILP).
// Result written into the h_seq region of d_out (consumed+overwritten by k2).
// -------------------------------------------------------------------------
__global__ __launch_bounds__(256) void xw_gemm_kernel(
    const float* __restrict__ x, const float* __restrict__ W,
    const float* __restrict__ bW, float* __restrict__ out)
{
  __shared__ __align__(16) float As[16 * LDA];
  const int tid = threadIdx.x;
  const long m0 = (long)blockIdx.x * 16;

  // Cooperative load of the 16x128 x tile (coalesced: rows are contiguous).
  for (int idx = tid; idx < 16 * I_DIM; idx += 256) {
    const int m = idx >> 7, k = idx & 127;
    As[m * LDA + k] = x[(m0 + m) * I_DIM + k];
  }
  __syncthreads();

  const int wave = tid >> 5, lane = tid & 31;
  const int lrow = lane & 15;          // A row / B col / C col within tile
  const int hi   = lane >> 4;          // half-wave select
  const int koff = hi << 1;            // K sub-offset for A/B fragments

  for (int sub = 0; sub < 2; ++sub) {
    const int n = wave * 32 + sub * 16 + lrow;   // global output column
    const float bias = bW[n];
    v8f acc;
#pragma unroll
    for (int j = 0; j < 8; ++j) acc[j] = bias;

#pragma unroll
    for (int s = 0; s < 32; ++s) {               // K = 128 = 32 * 4
      v2f a = *(const v2f*)(&As[lrow * LDA + 4 * s + koff]);        // x frag
      v2f b = *(const v2f*)(&W[(size_t)n * I_DIM + 4 * s + koff]);  // W^T frag
      acc = __builtin_amdgcn_wmma_f32_16x16x4_f32(
                false, a, false, b, (short)0, acc, false, false);
    }

#pragma unroll
    for (int j = 0; j < 8; ++j) {
      const int m = j + (hi << 3);               // C layout: M=j / j+8
      out[(m0 + m) * (size_t)H_DIM + n] = acc[j];
    }
  }
}

// -------------------------------------------------------------------------
// Kernel 2: sequential scan  h = tanh(xw_t + h @ V^T + b_V)
// 16 blocks, each owns 16 batch rows for all T steps (rows are independent).
// 16 waves / block; wave w owns output columns [16w, 16w+16) and keeps its
// K=256 x N=16 slab of V^T resident in 128 VGPRs for the whole kernel.
// h is double-buffered in LDS. Per step:
//   - xw tile for t+1 prefetched before the WMMA chain (latency hidden)
//   - split barrier: signal after LDS h stores, h_seq global stores issue
//     inside the barrier window, then wait.
// -------------------------------------------------------------------------
__global__ __launch_bounds__(512) void rnn_scan_kernel(
    const float* __restrict__ V, const float* __restrict__ bV,
    float* __restrict__ out)
{
  __shared__ __align__(16) float hbuf[2][16 * LDH];

  const int tid  = threadIdx.x;
  const int wave = tid >> 5, lane = tid & 31;
  const int lrow = lane & 15;
  const int hi   = lane >> 4;
  const int koff = hi << 1;
  const int n    = (wave << 4) + lrow;     // global output column (0..255)
  const int b0   = blockIdx.x << 4;        // batch-row base for this block

  // Preload V^T fragments: B[k][n] = V[n][k].  vt[s] covers K = 4s..4s+3.
  v2f vt[64];
#pragma unroll
  for (int s = 0; s < 64; ++s)
    vt[s] = *(const v2f*)(&V[(size_t)n * H_DIM + 4 * s + koff]);

  const float bias = bV[n];

  // h_{-1} = 0 lives in buffer 1 (t=0 uses prev = 1).
  for (int idx = tid; idx < 16 * LDH; idx += 512) hbuf[1][idx] = 0.f;

  // Prefetch xw tile for t = 0.
  float xw[8];
#pragma unroll
  for (int j = 0; j < 8; ++j) {
    const int m = j + (hi << 3);
    xw[j] = out[((size_t)b0 + m) * H_DIM + n];
  }
  __syncthreads();

  float hn[8];
#pragma unroll 1
  for (int t = 0; t < T_DIM; ++t) {
    const int cur = t & 1, prev = cur ^ 1;
    float* __restrict__ outt = out + ((size_t)t * B_DIM + b0) * H_DIM;

    // C init = xw_t (prefetched last step) + bias.
    v8f acc;
#pragma unroll
    for (int j = 0; j < 8; ++j) acc[j] = xw[j] + bias;

    // Prefetch next step's xw tile; latency overlaps the WMMA chain below.
    float xwn[8];
    if (t + 1 < T_DIM) {
      const float* __restrict__ outn = outt + (size_t)B_DIM * H_DIM;
#pragma unroll
      for (int j = 0; j < 8; ++j) {
        const int m = j + (hi << 3);
        xwn[j] = outn[m * H_DIM + n];
      }
    }

    // acc += h_prev @ V^T  (K = 256 = 64 * 4), V^T resident in VGPRs.
#pragma unroll
    for (int s = 0; s < 64; ++s) {
      v2f a = *(const v2f*)(&hbuf[prev][lrow * LDH + 4 * s + koff]);
      acc = __builtin_amdgcn_wmma_f32_16x16x4_f32(
                false, a, false, vt[s], (short)0, acc, false, false);
    }

    // tanh (hardware v_tanh_f32) + feed next step via LDS.
#pragma unroll
    for (int j = 0; j < 8; ++j) {
      const int m = j + (hi << 3);
      const float h = fast_tanh(acc[j]);
      hn[j] = h;
      hbuf[cur][m * LDH + n] = h;
    }

    // Split barrier: LDS h visible -> signal; h_seq global stores issue
    // inside the barrier window; then wait for all waves.
    asm volatile("s_wait_dscnt 0\n\ts_barrier_signal -1" ::: "memory");
#pragma unroll
    for (int j = 0; j < 8; ++j) {
      const int m = j + (hi << 3);
      outt[m * H_DIM + n] = hn[j];
    }
    asm volatile("s_barrier_wait -1" ::: "memory");

#pragma unroll
    for (int j = 0; j < 8; ++j) xw[j] = xwn[j];
  }

  // h_final
  float* __restrict__ hf = out + (size_t)T_DIM * B_DIM * H_DIM;
#pragma unroll
  for (int j = 0; j < 8; ++j) {
    const int m = j + (hi << 3);
    hf[(b0 + m) * H_DIM + n] = hn[j];
  }
}

extern "C" void kernel_launch(void* const* d_in, const int* in_sizes, int n_in,
                              void* d_out, int out_size, void* d_ws, size_t ws_size,
                              hipStream_t stream) {
  (void)in_sizes; (void)n_in; (void)out_size; (void)d_ws; (void)ws_size;
  const float* x  = (const float*)d_in[0];
  const float* W  = (const float*)d_in[1];
  const float* bW = (const float*)d_in[2];
  const float* V  = (const float*)d_in[3];
  const float* bV = (const float*)d_in[4];
  float* out = (float*)d_out;

  // Phase 1: fully parallel input projection into the h_seq region.
  xw_gemm_kernel<<<(T_DIM * B_DIM) / 16, 256, 0, stream>>>(x, W, bW, out);
  // Phase 2: per-batch-chunk sequential scan (no cross-block deps).
  rnn_scan_kernel<<<B_DIM / 16, 512, 0, stream>>>(V, bV, out);
}